// GVAE_30236569764092
// MI455X (gfx1250) — compile-verified
//
#include <hip/hip_runtime.h>
#include <hip/hip_bf16.h>

#define HIDDEN 32
#define LATENT 8
#define DEC_N  100

typedef __attribute__((ext_vector_type(2))) float v2f;
typedef __attribute__((ext_vector_type(8))) float v8f;

// ---------------------------------------------------------------------------
// Phase 1: graph preprocessing (memory / atomic bound, L2-resident node data)
// ---------------------------------------------------------------------------

// deg starts at 1.0 (the self-loop contributes exactly 1 to every node).
__global__ void gvae_init_deg(float* __restrict__ deg, int N) {
    int i = blockIdx.x * blockDim.x + threadIdx.x;
    if (i < N) deg[i] = 1.0f;
}

__global__ void gvae_init_pool(float* __restrict__ psum, float* __restrict__ cnt, int G) {
    int i = blockIdx.x * blockDim.x + threadIdx.x;
    if (i < G * HIDDEN)            psum[i] = 0.0f;
    else if (i < G * (HIDDEN + 1)) cnt[i - G * HIDDEN] = 0.0f;
}

__global__ void gvae_degree(const int* __restrict__ ei, float* __restrict__ deg, int E) {
    int e = blockIdx.x * blockDim.x + threadIdx.x;
    if (e < E) atomicAdd(&deg[ei[E + e]], 1.0f);  // dst = ei[1][e]
}

// dinv = rsqrt(deg) (deg >= 1 always); seed s with the self-loop message.
__global__ void gvae_dinv_selfloop(const float* __restrict__ x,
                                   float* __restrict__ deg_dinv,
                                   float* __restrict__ s, int N) {
    int i = blockIdx.x * blockDim.x + threadIdx.x;
    if (i < N) {
        float dinv = rsqrtf(deg_dinv[i]);
        deg_dinv[i] = dinv;
        s[i] = x[i] * dinv * dinv;
    }
}

// s[dst] += x[src] * dinv[src] * dinv[dst]   (the whole GCN, collapsed to scalar)
__global__ void gvae_edge_msg(const int* __restrict__ ei, const float* __restrict__ x,
                              const float* __restrict__ dinv, float* __restrict__ s, int E) {
    int e = blockIdx.x * blockDim.x + threadIdx.x;
    if (e < E) {
        int src = ei[e];
        int dst = ei[E + e];
        atomicAdd(&s[dst], x[src] * dinv[src] * dinv[dst]);
    }
}

// One wave per node: lane k owns hidden dim k -> coalesced atomics into psum.
__global__ void gvae_pool(const float* __restrict__ s, const int* __restrict__ batch,
                          const float* __restrict__ W, const float* __restrict__ b,
                          float* __restrict__ psum, float* __restrict__ cnt, int N) {
    int idx = blockIdx.x * blockDim.x + threadIdx.x;
    int n = idx >> 5;
    int k = idx & 31;
    if (n < N) {
        float h = fmaf(s[n], W[k], b[k]);
        h = h > 0.0f ? h : 0.0f;                 // relu
        int g = batch[n];
        atomicAdd(&psum[g * HIDDEN + k], h);
        if (k == 0) atomicAdd(&cnt[g], 1.0f);
    }
}

__global__ void gvae_finalize_pool(const float* __restrict__ psum, const float* __restrict__ cnt,
                                   float* __restrict__ pooled, int G) {
    int i = blockIdx.x * blockDim.x + threadIdx.x;
    if (i < G * HIDDEN) pooled[i] = psum[i] / fmaxf(cnt[i >> 5], 1.0f);
}

// ---------------------------------------------------------------------------
// Phase 2: encoder GEMM on the matrix pipe.
// A = pooled [G x 32], B = [w_mu | w_lv] packed as 32x16.
// One wave per 16-row tile, 8 chained V_WMMA_F32_16X16X4_F32 ops (K=32).
// C/D layout: VGPR i, lanes 0-15 -> (M=i, N=lane); lanes 16-31 -> (M=i+8, N=lane-16).
// A 16x4 layout: lanes 0-15 hold K={k0,k0+1}, lanes 16-31 hold K={k0+2,k0+3}.
// ---------------------------------------------------------------------------
__global__ __launch_bounds__(32)
void gvae_encoder_wmma(const float* __restrict__ pooled,
                       const float* __restrict__ w_mu, const float* __restrict__ b_mu,
                       const float* __restrict__ w_lv, const float* __restrict__ b_lv,
                       const float* __restrict__ eps,
                       float* __restrict__ mu_out, float* __restrict__ lv_out,
                       float* __restrict__ z) {
    const int lane = threadIdx.x;
    const int base = blockIdx.x * 16;
    const int n    = lane & 15;          // output column within the 16-wide tile
    const bool hi  = lane >= 16;
    const int arow = base + (hi ? lane - 16 : lane);
    // column n<8 -> w_mu[:,n]; n>=8 -> w_lv[:,n-8]
    const float* __restrict__ bcol = (n < 8) ? (w_mu + n) : (w_lv + (n - 8));

    v8f c = {};
#pragma unroll
    for (int kb = 0; kb < HIDDEN / 4; ++kb) {
        const int k0 = kb * 4 + (hi ? 2 : 0);
        v2f a, bb;
        const float* ap = pooled + arow * HIDDEN + k0;
        a.x  = ap[0];
        a.y  = ap[1];
        bb.x = bcol[(k0 + 0) * LATENT];
        bb.y = bcol[(k0 + 1) * LATENT];
        c = __builtin_amdgcn_wmma_f32_16x16x4_f32(false, a, false, bb, (short)0, c,
                                                  false, false);
    }

#pragma unroll
    for (int i = 0; i < 8; ++i) {
        const int m = base + i + (hi ? 8 : 0);
        float v = c[i] + ((n < 8) ? b_mu[n] : b_lv[n - 8]);
        if (n < 8) mu_out[m * LATENT + n]       = v;
        else       lv_out[m * LATENT + (n - 8)] = v;
        // lane j holds mu[:,j]; lane j+8 holds logvar[:,j] for the SAME rows
        float partner = __shfl_xor(v, 8, 32);
        if (n < 8) z[m * LATENT + n] = fmaf(eps[m * LATENT + n], expf(0.5f * partner), v);
    }
}

// ---------------------------------------------------------------------------
// Phase 3: decoder GEMM + hard sigmoid threshold. A = z [G x 8], B = w_dec [8 x 100].
// Grid (G/16, 7) waves; 2 WMMA k-steps each; masked loads for cols >= 100.
// ---------------------------------------------------------------------------
__global__ __launch_bounds__(32)
void gvae_decoder_wmma(const float* __restrict__ z, const float* __restrict__ w_dec,
                       const float* __restrict__ b_dec, float* __restrict__ adj) {
    const int lane = threadIdx.x;
    const int base = blockIdx.x * 16;
    const int n    = lane & 15;
    const bool hi  = lane >= 16;
    const int col  = blockIdx.y * 16 + n;
    const int colc = col < DEC_N ? col : (DEC_N - 1);   // clamp address
    const float cm = col < DEC_N ? 1.0f : 0.0f;         // multiplicative mask
    const int arow = base + (hi ? lane - 16 : lane);

    v8f c = {};
#pragma unroll
    for (int kb = 0; kb < LATENT / 4; ++kb) {
        const int k0 = kb * 4 + (hi ? 2 : 0);
        v2f a, bb;
        const float* ap = z + arow * LATENT + k0;
        a.x  = ap[0];
        a.y  = ap[1];
        bb.x = w_dec[(k0 + 0) * DEC_N + colc] * cm;
        bb.y = w_dec[(k0 + 1) * DEC_N + colc] * cm;
        c = __builtin_amdgcn_wmma_f32_16x16x4_f32(false, a, false, bb, (short)0, c,
                                                  false, false);
    }

    const float bias = b_dec[colc];
#pragma unroll
    for (int i = 0; i < 8; ++i) {
        const int m = base + i + (hi ? 8 : 0);
        float logit = c[i] + bias;
        if (col < DEC_N) adj[m * DEC_N + col] = (logit > 0.0f) ? 1.0f : 0.0f;
    }
}

// ---------------------------------------------------------------------------

extern "C" void kernel_launch(void* const* d_in, const int* in_sizes, int n_in,
                              void* d_out, int out_size, void* d_ws, size_t ws_size,
                              hipStream_t stream) {
    const float* x     = (const float*)d_in[0];
    const int*   ei    = (const int*)  d_in[1];
    const int*   batch = (const int*)  d_in[2];
    const float* W     = (const float*)d_in[3];
    const float* b     = (const float*)d_in[4];
    const float* w_mu  = (const float*)d_in[5];
    const float* b_mu  = (const float*)d_in[6];
    const float* w_lv  = (const float*)d_in[7];
    const float* b_lv  = (const float*)d_in[8];
    const float* w_dec = (const float*)d_in[9];
    const float* b_dec = (const float*)d_in[10];
    const float* eps   = (const float*)d_in[11];

    const int N = in_sizes[0];          // 100000
    const int E = in_sizes[1] / 2;      // 2000000
    const int G = in_sizes[11] / LATENT; // 1024

    // Workspace carve-up (floats)
    float* ws       = (float*)d_ws;
    float* deg_dinv = ws;                       // [N]
    float* s        = ws + N;                   // [N]
    float* psum     = ws + 2 * N;               // [G*32]
    float* cnt      = psum + G * HIDDEN;        // [G]
    float* pooled   = cnt + G;                  // [G*32]
    float* z        = pooled + G * HIDDEN;      // [G*8]

    // Output carve-up: adj [G*100] | mu [G*8] | logvar [G*8]
    float* adj    = (float*)d_out;
    float* mu_out = adj + G * DEC_N;
    float* lv_out = mu_out + G * LATENT;

    const int TB = 256;
    gvae_init_deg<<<(N + TB - 1) / TB, TB, 0, stream>>>(deg_dinv, N);
    gvae_init_pool<<<(G * (HIDDEN + 1) + TB - 1) / TB, TB, 0, stream>>>(psum, cnt, G);
    gvae_degree<<<(E + TB - 1) / TB, TB, 0, stream>>>(ei, deg_dinv, E);
    gvae_dinv_selfloop<<<(N + TB - 1) / TB, TB, 0, stream>>>(x, deg_dinv, s, N);
    gvae_edge_msg<<<(E + TB - 1) / TB, TB, 0, stream>>>(ei, x, deg_dinv, s, E);
    gvae_pool<<<(int)(((long)N * 32 + TB - 1) / TB), TB, 0, stream>>>(s, batch, W, b,
                                                                      psum, cnt, N);
    gvae_finalize_pool<<<(G * HIDDEN + TB - 1) / TB, TB, 0, stream>>>(psum, cnt, pooled, G);
    gvae_encoder_wmma<<<G / 16, 32, 0, stream>>>(pooled, w_mu, b_mu, w_lv, b_lv, eps,
                                                 mu_out, lv_out, z);
    gvae_decoder_wmma<<<dim3(G / 16, (DEC_N + 15) / 16), 32, 0, stream>>>(z, w_dec,
                                                                          b_dec, adj);
}